// GravityDecoder_5162550690701
// MI455X (gfx1250) — compile-verified
//
#include <hip/hip_runtime.h>

// GravityDecoder sampled-adjacency kernel for MI455X (gfx1250, wave32).
//
// Strategy: gather-based per-edge evaluation (never materialize the N×N adj).
// One wave handles 16 edges: the 16 src rows form A (16x128), the 16 dst rows
// form B (128x16); 32 chained v_wmma_f32_16x16x4_f32 accumulate C = A*B and
// the diagonal of C gives the 16 edge dot products. Row norms are accumulated
// from the same loaded elements (each lane covers half a row; lane L pairs
// with lane L^16 via shfl_xor). All data stays f32 to match the reference's
// log/sigmoid sensitivity; whole z (8.45 MB) is L2-resident (192 MB).

typedef __attribute__((ext_vector_type(2))) float v2f;
typedef __attribute__((ext_vector_type(8))) float v8f;

#define DFULL 129   // 128 features + mass
#define DR    128
#define GD_EPS 0.01f

__device__ __forceinline__ float fast_sigmoid(float x) {
    return 1.0f / (1.0f + __expf(-x));
}

__global__ __launch_bounds__(256) void gravity_decoder_wmma_kernel(
    const float* __restrict__ z,          // (N, 129)
    const float* __restrict__ lscale,     // (1,)
    const long long* __restrict__ eli,    // (2, E) int64
    float* __restrict__ out,              // (E,)
    int E)
{
    const int lane  = threadIdx.x & 31;
    const int wave  = threadIdx.x >> 5;
    const int group = blockIdx.x * 8 + wave;   // 16 edges per wave
    const int base  = group * 16;
    const int eSlot = lane & 15;               // which edge-of-16 this lane serves

    // Edge endpoints for this lane's slot (i = src row, j = dst row).
    const int aRow = (int)eli[base + eSlot];
    const int bRow = (int)eli[(long)E + base + eSlot];

    // f32 WMMA 16x16x4 operand layout:
    //   A (16x4, MxK): lanes 0-15 -> M=lane, VGPR{0,1} = K{k,k+1};
    //                  lanes 16-31 -> M=lane-16, VGPR{0,1} = K{k+2,k+3}.
    //   B (4x16, KxN): mirrored (N across lanes, K across VGPRs/half-wave).
    const int kHalf = (lane >> 4) << 1;        // 0 for lanes 0-15, 2 for 16-31
    const float* __restrict__ ap = z + (long)aRow * DFULL + kHalf;
    const float* __restrict__ bp = z + (long)bRow * DFULL + kHalf;

    v8f  c  = {0.f, 0.f, 0.f, 0.f, 0.f, 0.f, 0.f, 0.f};
    float pa = 0.f, pb = 0.f;                  // partial row norms (half-row each)

    #pragma unroll 4
    for (int k = 0; k < DR; k += 4) {
        const float a0 = ap[k], a1 = ap[k + 1];
        const float b0 = bp[k], b1 = bp[k + 1];
        v2f a = {a0, a1};
        v2f b = {b0, b1};
        // D = A*B + C  (f32, 16x16x4); uniform control flow -> EXEC all ones.
        c = __builtin_amdgcn_wmma_f32_16x16x4_f32(
                /*neg_a=*/false, a, /*neg_b=*/false, b,
                /*c_mod=*/(short)0, c, /*reuse_a=*/false, /*reuse_b=*/false);
        pa = __builtin_fmaf(a0, a0, pa); pa = __builtin_fmaf(a1, a1, pa);
        pb = __builtin_fmaf(b0, b0, pb); pb = __builtin_fmaf(b1, b1, pb);
    }

    // Combine half-row norms: lane L and lane L^16 cover complementary K's
    // of the same row, so both end up holding the full ||r||^2.
    const float normA = pa + __shfl_xor(pa, 16, 32);
    const float normB = pb + __shfl_xor(pb, 16, 32);

    // Diagonal extraction from C (32-bit 16x16 layout):
    //   edge e (e<8):  lane e,     VGPR e
    //   edge e (e>=8): lane e+16,  VGPR e-8
    const bool ownerLo = (lane < 8);
    const bool ownerHi = (lane >= 24);
    if (ownerLo || ownerHi) {
        const int v = ownerLo ? lane : (lane - 24);
        const float dot = c[v];

        const float r2   = normA + normB - 2.0f * dot;
        const float mass = z[(long)bRow * DFULL + DR];   // mass of dst row j
        const float t    = mass - lscale[0] * __logf(r2 + GD_EPS);
        const float adj  = fast_sigmoid(t);

        const int e = ownerLo ? lane : (lane - 16);
        out[base + e] = fast_sigmoid(adj);
    }
}

extern "C" void kernel_launch(void* const* d_in, const int* in_sizes, int n_in,
                              void* d_out, int out_size, void* d_ws, size_t ws_size,
                              hipStream_t stream) {
    const float*     z   = (const float*)d_in[0];
    const float*     l   = (const float*)d_in[1];
    const long long* eli = (const long long*)d_in[2];
    float*           out = (float*)d_out;

    const int E = in_sizes[2] / 2;        // eli is (2, E)
    const int groups = E / 16;            // 16 edges per wave
    const int blocks = (groups + 7) / 8;  // 8 waves (256 threads) per block

    gravity_decoder_wmma_kernel<<<blocks, 256, 0, stream>>>(z, l, eli, out, E);
}